// AngularTripletCenterLoss_37580963840391
// MI455X (gfx1250) — compile-verified
//
#include <hip/hip_runtime.h>
#include <math.h>

// AngularTripletCenterLoss, S=1024 speakers, U=128 utts, D=64 feat dim.
// Stage1: per-speaker centroid/normalize/intra (one streaming pass over x, 32MB).
// Stage2: sel @ cent^T via V_WMMA_F32_16X16X4_F32 (fp32 matrix path) + fused
//         off-diagonal row-max, arccos, hinge.
// Stage3: scalar reduction.

#define SPK 1024
#define UTT 128
#define DIM 64
#define MARGIN 0.5f
#define CEPS 1e-7f
#define NEG_INF (-3.402823466e38f)

typedef __attribute__((ext_vector_type(2))) float v2f;
typedef __attribute__((ext_vector_type(8))) float v8f;

// ---------------- Stage 1: centroids, intra angles, hardest utterance ------
__global__ __launch_bounds__(256) void atc_stage1(const float* __restrict__ x,
                                                  float* __restrict__ cent,
                                                  float* __restrict__ sel,
                                                  float* __restrict__ intraD)
{
    __shared__ float xs[UTT * DIM];   // 32 KB speaker tile
    __shared__ float cn[DIM];
    __shared__ float cosArr[UTT];
    __shared__ float s_inv;
    __shared__ int   s_idx;

    const int s   = blockIdx.x;
    const int tid = threadIdx.x;

    // Stream x[s] (128x64 fp32) into LDS with float4 loads (coalesced).
    const float4* xg  = (const float4*)(x + (size_t)s * UTT * DIM);
    float4*       xls = (float4*)xs;
    for (int i = tid; i < UTT * DIM / 4; i += 256) xls[i] = xg[i];
    __syncthreads();

    // Centroid: thread d sums column d (bank = d per lane -> conflict-free).
    if (tid < DIM) {
        float acc = 0.0f;
        for (int u = 0; u < UTT; ++u) acc += xs[u * DIM + tid];
        cn[tid] = acc * (1.0f / UTT);
    }
    __syncthreads();
    if (tid == 0) {
        float ss = 0.0f;
        for (int d = 0; d < DIM; ++d) ss += cn[d] * cn[d];
        s_inv = 1.0f / fmaxf(sqrtf(ss), 1e-12f);   // F.normalize eps
    }
    __syncthreads();
    if (tid < DIM) {
        float v = cn[tid] * s_inv;
        cn[tid] = v;
        cent[s * DIM + tid] = v;
    }
    __syncthreads();

    // Intra cos: thread u dots raw utterance u with normalized centroid.
    // Rotated index (i+u)&63 keeps both LDS reads bank-conflict-free.
    if (tid < UTT) {
        float dot = 0.0f;
        for (int i = 0; i < DIM; ++i) {
            int d = (i + tid) & (DIM - 1);
            dot += xs[(tid << 6) + d] * cn[d];
        }
        cosArr[tid] = fminf(fmaxf(dot, -1.0f + CEPS), 1.0f - CEPS);
    }
    __syncthreads();

    // First-occurrence argmin of clipped cos == jnp.argmax of arccos(clip).
    if (tid == 0) {
        float best = cosArr[0];
        int   bi   = 0;
        for (int u = 1; u < UTT; ++u) {
            float v = cosArr[u];
            if (v < best) { best = v; bi = u; }
        }
        s_idx = bi;
        intraD[s] = acosf(best);      // max intra angle
    }
    __syncthreads();
    if (tid < DIM) sel[s * DIM + tid] = xs[(s_idx << 6) + tid];
}

// -------- Stage 2: dots = sel @ cent^T (WMMA f32), off-diag row-max --------
__global__ __launch_bounds__(256) void atc_stage2(const float* __restrict__ sel,
                                                  const float* __restrict__ cent,
                                                  const float* __restrict__ intraD,
                                                  float* __restrict__ rowloss)
{
    __shared__ float rmax[8 * 16];    // per-wave partial row maxima

    const int tid  = threadIdx.x;
    const int lane = tid & 31;
    const int w    = tid >> 5;        // 8 waves per block
    const int m0   = blockIdx.x * 16; // 16-row strip
    const int r16  = lane & 15;
    const int hi   = lane >> 4;       // half-wave: 0 or 1
    const int kb   = hi * 2;          // K sub-offset per ISA A/B layout

    // A fragments for the whole K=64: 16 steps of K=4.
    // ISA 16x4 f32 A layout: lane<16 -> M=lane, VGPR0/1 = K={0,1};
    //                        lane>=16 -> M=lane-16, K={2,3}.  => float2 load.
    v2f a[16];
    {
        const float* ap = sel + (m0 + r16) * DIM + kb;
        #pragma unroll
        for (int kk = 0; kk < 16; ++kk)
            a[kk] = *(const v2f*)(ap + kk * 4);
    }

    float rm[8];
    #pragma unroll
    for (int r = 0; r < 8; ++r) rm[r] = NEG_INF;

    // Each wave covers 8 column tiles -> 128 columns; 8 waves -> all 1024.
    for (int t = 0; t < 8; ++t) {
        const int n0 = (w * 8 + t) * 16;
        // B 4x16 layout mirrors A: lane<16 -> N=lane, K={0,1}; lane>=16 K={2,3}.
        // B[k][n] = cent[n0+n][k]  => contiguous float2 in cent row.
        const float* bp = cent + (n0 + r16) * DIM + kb;
        v8f c = {};
        #pragma unroll
        for (int kk = 0; kk < 16; ++kk) {
            v2f b = *(const v2f*)(bp + kk * 4);
            c = __builtin_amdgcn_wmma_f32_16x16x4_f32(
                    false, a[kk], false, b, (short)0, c, false, false);
        }
        // C layout: VGPR r, lane<16 -> (M=r, N=lane); lane>=16 -> (M=8+r, N=lane-16)
        const int col = n0 + r16;
        #pragma unroll
        for (int r = 0; r < 8; ++r) {
            const int row = m0 + r + hi * 8;
            float v = (col == row) ? NEG_INF : c[r];   // drop diagonal
            rm[r] = fmaxf(rm[r], v);
        }
    }

    // Butterfly max across the 16 lanes of each half-wave.
    #pragma unroll
    for (int r = 0; r < 8; ++r) {
        float v = rm[r];
        v = fmaxf(v, __shfl_xor(v, 1, 32));
        v = fmaxf(v, __shfl_xor(v, 2, 32));
        v = fmaxf(v, __shfl_xor(v, 4, 32));
        v = fmaxf(v, __shfl_xor(v, 8, 32));
        if (r16 == 0) rmax[w * 16 + hi * 8 + r] = v;
    }
    __syncthreads();

    // Merge the 8 waves; min inter angle = arccos(clip(max cos)); hinge loss.
    if (tid < 16) {
        float m = NEG_INF;
        for (int ww = 0; ww < 8; ++ww) m = fmaxf(m, rmax[ww * 16 + tid]);
        float cc     = fminf(fmaxf(m, -1.0f + CEPS), 1.0f - CEPS);
        float interd = acosf(cc);
        float l      = MARGIN + intraD[m0 + tid] - interd;
        rowloss[m0 + tid] = fmaxf(l, 0.0f);
    }
}

// ---------------- Stage 3: 1024 -> 1 reduction -----------------------------
__global__ __launch_bounds__(256) void atc_stage3(const float* __restrict__ rowloss,
                                                  float* __restrict__ out)
{
    __shared__ float red[256];
    const int tid = threadIdx.x;
    float acc = 0.0f;
    for (int i = tid; i < SPK; i += 256) acc += rowloss[i];
    red[tid] = acc;
    __syncthreads();
    for (int st = 128; st > 0; st >>= 1) {
        if (tid < st) red[tid] += red[tid + st];
        __syncthreads();
    }
    if (tid == 0) out[0] = red[0];
}

extern "C" void kernel_launch(void* const* d_in, const int* in_sizes, int n_in,
                              void* d_out, int out_size, void* d_ws, size_t ws_size,
                              hipStream_t stream) {
    const float* x = (const float*)d_in[0];   // [1024,128,64] fp32

    // Workspace layout (floats): cent | sel | intraD | rowloss  (~520 KB)
    float* ws      = (float*)d_ws;
    float* cent    = ws;                       // SPK*DIM
    float* sel     = cent   + SPK * DIM;       // SPK*DIM
    float* intraD  = sel    + SPK * DIM;       // SPK
    float* rowloss = intraD + SPK;             // SPK
    float* out     = (float*)d_out;            // scalar fp32

    atc_stage1<<<SPK,      256, 0, stream>>>(x, cent, sel, intraD);
    atc_stage2<<<SPK / 16, 256, 0, stream>>>(sel, cent, intraD, rowloss);
    atc_stage3<<<1,        256, 0, stream>>>(rowloss, out);
}